// MetScore_46677704573703
// MI455X (gfx1250) — compile-verified
//
#include <hip/hip_runtime.h>
#include <math.h>

// ---------------------------------------------------------------------------
// MetScore for MI455X (gfx1250, wave32).
// Stage 1: streaming per-plane partial reduction (memory-bound, ~166MB @ 23.3TB/s).
//          Packed f16 threshold indicators (v_pk_fma) + v_fma_mix accumulation
//          keep the per-element VALU cost near the HBM roofline.
// Stage 2: one block, 20 waves; per-timestep reduction of 128 partials x 32
//          accumulators done with v_wmma_f32_16x16x4_f32 (A = ones), then the
//          scalar score math. Fully deterministic (no atomics).
// ---------------------------------------------------------------------------

typedef _Float16 half2_t __attribute__((ext_vector_type(2)));
typedef float    v2f     __attribute__((ext_vector_type(2)));
typedef float    v8f     __attribute__((ext_vector_type(8)));

#define HW     230400          // 480*480 elements per (b,t) plane
#define NQ     57600           // float4 per plane
#define KB     32              // blocks per plane
#define CHUNK  1800            // NQ / KB
#define TPB    256
#define NACC   32              // padded accumulator count (27 used)
#define NT     20
#define NB     4
#define NBLK   (NB * NT * KB)  // 2560

// ---------------------------------------------------------------------------
// Per-element update. Indicators computed in packed f16:
//   ind_l = clamp((x - T_l) * 2^14 + 1, 0, 1)  -> exactly 0 or 1 per half.
// x holds (p', t') where p' = mask ? p : -1 (folds mask into the comparisons).
// ---------------------------------------------------------------------------
__device__ __forceinline__ void met_elem(float pr, float tg, bool m,
    float& n_c, float& s_p, float& s_t, float& s_pp, float& s_tt, float& s_pt,
    float& n_v, float* hits, half2_t* bins, float* maen)
{
  const float thr[5] = {0.1f, 1.0f, 2.0f, 5.0f, 8.0f};
  float p = pr * 30.0f;
  float t = tg * 30.0f;
  float pm_in = m ? p : -1.0f;
  float tm_in = m ? t : -1.0f;

  half2_t x;
  x.x = (_Float16)pm_in;
  x.y = (_Float16)tm_in;

  const half2_t K    = {(_Float16)16384.0f, (_Float16)16384.0f}; // 2^14
  const half2_t one  = {(_Float16)1.0f,     (_Float16)1.0f};
  const half2_t zero = {(_Float16)0.0f,     (_Float16)0.0f};

  half2_t ind[6];
#pragma unroll
  for (int l = 0; l < 5; ++l) {
    half2_t T2 = {(_Float16)thr[l], (_Float16)thr[l]};
    half2_t d  = x - T2;              // sign exact (Sterbenz near thresholds)
    half2_t w  = d * K + one;         // v_pk_fma_f16
    w = __builtin_elementwise_max(w, zero);
    w = __builtin_elementwise_min(w, one);
    ind[l] = w;                       // exactly 0 or 1 per half
  }
  ind[5] = zero;

  float ad = fabsf(p - t);
#pragma unroll
  for (int l = 0; l < 5; ++l) {
    half2_t pin = ind[l] - ind[l + 1];            // {p in level, t in level}
    bins[l] += pin;                               // pcnt (lo), cnt (hi), exact f16 ints
    hits[l] += (float)pin.x * (float)pin.y;       // v_fma_mix_f32
    maen[l] += (float)pin.y * ad;                 // v_fma_mix_f32
  }

  // mc = mask & !(p<0.1 & t<0.1)  ==  (p'>=0.1) | (t'>=0.1)
  float oc = (float)ind[0].x + (float)ind[0].y;
  float mc = fminf(oc, 1.0f);
  n_c += mc;
  float pv = p * mc;
  float tv = t * mc;
  s_p += pv;
  s_t += tv;
  s_pp = fmaf(p, pv, s_pp);
  s_tt = fmaf(t, tv, s_tt);
  s_pt = fmaf(p, tv, s_pt);
  n_v += m ? 1.0f : 0.0f;
}

// ---------------------------------------------------------------------------
// Stage 1: each block handles 1/KB of one (b,t) plane; writes 32 partial sums.
// ---------------------------------------------------------------------------
__global__ __launch_bounds__(TPB) void met_partial(
    const float* __restrict__ pred, const float* __restrict__ targ,
    const unsigned char* __restrict__ mask, float* __restrict__ part)
{
  int blk   = blockIdx.x;        // 0..NBLK-1
  int plane = blk / KB;          // b*20 + t
  int kb    = blk % KB;
  int tid   = threadIdx.x;

  long base = (long)plane * HW;
  const float4*       p4 = (const float4*)(pred + base);
  const float4*       t4 = (const float4*)(targ + base);
  const unsigned int* m4 = (const unsigned int*)(mask + base);

  float n_c = 0.f, s_p = 0.f, s_t = 0.f, s_pp = 0.f, s_tt = 0.f, s_pt = 0.f, n_v = 0.f;
  float   hits[5] = {0.f, 0.f, 0.f, 0.f, 0.f};
  float   maen[5] = {0.f, 0.f, 0.f, 0.f, 0.f};
  half2_t bins[5];
#pragma unroll
  for (int l = 0; l < 5; ++l) { bins[l].x = (_Float16)0.0f; bins[l].y = (_Float16)0.0f; }

  int lo = kb * CHUNK, hi = lo + CHUNK;
  for (int i = lo + tid; i < hi; i += TPB) {
    float4 pv = p4[i];
    float4 tv = t4[i];
    unsigned int mm = m4[i];
    met_elem(pv.x, tv.x, (mm & 0x000000ffu) != 0u, n_c, s_p, s_t, s_pp, s_tt, s_pt, n_v, hits, bins, maen);
    met_elem(pv.y, tv.y, (mm & 0x0000ff00u) != 0u, n_c, s_p, s_t, s_pp, s_tt, s_pt, n_v, hits, bins, maen);
    met_elem(pv.z, tv.z, (mm & 0x00ff0000u) != 0u, n_c, s_p, s_t, s_pp, s_tt, s_pt, n_v, hits, bins, maen);
    met_elem(pv.w, tv.w, (mm & 0xff000000u) != 0u, n_c, s_p, s_t, s_pp, s_tt, s_pt, n_v, hits, bins, maen);
  }

  float acc[NACC];
  acc[0] = n_c; acc[1] = s_p; acc[2] = s_t; acc[3] = s_pp; acc[4] = s_tt; acc[5] = s_pt; acc[6] = n_v;
#pragma unroll
  for (int l = 0; l < 5; ++l) {
    acc[7  + l] = hits[l];
    acc[12 + l] = (float)bins[l].x;   // pcnt  (pred in level, masked)
    acc[17 + l] = (float)bins[l].y;   // cnt   (target in level, masked)
    acc[22 + l] = maen[l];
  }
#pragma unroll
  for (int k = 27; k < NACC; ++k) acc[k] = 0.f;

  // Deterministic LDS tree reduction: 256 rows x 32 floats.
  __shared__ float4 sh[TPB * 8];
#pragma unroll
  for (int q = 0; q < 8; ++q)
    sh[tid * 8 + q] = make_float4(acc[4 * q], acc[4 * q + 1], acc[4 * q + 2], acc[4 * q + 3]);
  __syncthreads();
#pragma unroll
  for (int s = 128; s >= 32; s >>= 1) {
    if (tid < s) {
#pragma unroll
      for (int q = 0; q < 8; ++q) {
        float4 a = sh[tid * 8 + q];
        float4 b = sh[(tid + s) * 8 + q];
        sh[tid * 8 + q] = make_float4(a.x + b.x, a.y + b.y, a.z + b.z, a.w + b.w);
      }
    }
    __syncthreads();
  }
  if (tid < NACC) {
    const float* shf = (const float*)sh;
    float v = 0.f;
#pragma unroll
    for (int r = 0; r < 32; ++r) v += shf[r * NACC + tid];
    part[blk * NACC + tid] = v;
  }
}

// ---------------------------------------------------------------------------
// Stage 2: one block, 20 waves (one per timestep). Each wave reduces its
// 128 partials x 32 accumulators with v_wmma_f32_16x16x4_f32 (A = ones so
// every D row is the column-sum of B), then threads 0..19 do the score math.
// ---------------------------------------------------------------------------
__global__ __launch_bounds__(NT * 32) void met_final(
    const float* __restrict__ part, float* __restrict__ out)
{
  __shared__ float red[NT][NACC];
  __shared__ float sh_score[NT];
  __shared__ float sh_ts[NT][5];
  __shared__ float sh_mae[NT][5];

  int tid  = threadIdx.x;
  int wave = tid >> 5;
  int lane = tid & 31;

  {
    int t  = wave;
    int n  = lane & 15;
    int hb = lane >> 4;               // 0: K rows 0/1, 1: K rows 2/3
    v2f a; a.x = 1.0f; a.y = 1.0f;    // ones A-matrix (16x4)
#pragma unroll
    for (int g = 0; g < 2; ++g) {     // accumulator halves 0..15, 16..31
      v8f c = {0.f, 0.f, 0.f, 0.f, 0.f, 0.f, 0.f, 0.f};
#pragma unroll
      for (int ch = 0; ch < 32; ++ch) {         // 128 partials / 4 per wmma
        int j0 = 4 * ch + 2 * hb;               // K = 0 or 2
        int j1 = j0 + 1;                        // K = 1 or 3
        int b0 = j0 >> 5, k0 = j0 & 31;
        int b1 = j1 >> 5, k1 = j1 & 31;
        int blk0 = (b0 * NT + t) * KB + k0;
        int blk1 = (b1 * NT + t) * KB + k1;
        v2f bm;
        bm.x = part[blk0 * NACC + g * 16 + n];
        bm.y = part[blk1 * NACC + g * 16 + n];
        c = __builtin_amdgcn_wmma_f32_16x16x4_f32(
                false, a, false, bm, (short)0, c, false, false);
      }
      if (lane < 16) red[t][g * 16 + lane] = c[0];   // D row M=0
    }
  }
  __syncthreads();

  const float LW[5] = {0.1f, 0.1f, 0.2f, 0.25f, 0.35f};
  const float TW[NT] = {0.0075f, 0.02f, 0.03f, 0.04f, 0.05f, 0.06f, 0.07f, 0.08f,
                        0.09f, 0.1f, 0.09f, 0.08f, 0.07f, 0.06f, 0.05f, 0.04f,
                        0.03f, 0.02f, 0.0075f, 0.005f};

  if (tid < NT) {
    int t = tid;
    const float* R = red[t];
    float n_c = R[0], s_p = R[1], s_t = R[2], s_pp = R[3], s_tt = R[4], s_pt = R[5], n_v = R[6];
    float safe = fmaxf(n_c, 1.0f);
    float pm = s_p / safe;
    float tm = s_t / safe;
    float num = s_pt - tm * s_p - pm * s_t + pm * tm * n_c;
    float dp2 = fmaxf(s_pp - 2.0f * pm * s_p + pm * pm * n_c, 0.0f);
    float dt2 = fmaxf(s_tt - 2.0f * tm * s_t + tm * tm * n_c, 0.0f);
    float den = sqrtf(dp2 * dt2);
    float r = fminf(fmaxf(num / (den + 1e-6f), -1.0f), 1.0f);
    r = (n_c > 0.0f) ? r : 0.0f;
    bool any_valid = (n_v > 0.0f);

    float term_corr = sqrtf(__expf(r - 1.0f));
    float sum_level = 0.0f;
#pragma unroll
    for (int l = 0; l < 5; ++l) {
      float h  = R[7 + l];
      float pc = R[12 + l];
      float cn = R[17 + l];
      float mn = R[22 + l];
      // hits+misses+fas = cn + pc - h
      float ts  = h / (cn + pc - h + 1e-8f);
      float mae = (cn > 0.0f) ? (mn / fmaxf(cn, 1.0f)) : 0.0f;
      if (!any_valid) { ts = 0.0f; mae = 0.0f; }
      sh_ts[t][l]  = ts;
      sh_mae[t][l] = mae;
      out[41  + t * 5 + l] = ts;
      out[141 + t * 5 + l] = mae;
      sum_level += LW[l] * ts * sqrtf(__expf(-mae * 0.01f));
    }
    float score = term_corr * sum_level;
    sh_score[t]  = score;
    out[1  + t] = score;
    out[21 + t] = r;
  }
  __syncthreads();

  if (tid == 0) {
    float total = 0.0f;
#pragma unroll
    for (int t = 0; t < NT; ++t) total += sh_score[t] * TW[t];
    out[0] = total;
  }
  if (tid >= 1 && tid < 11) {            // 5 ts-means + 5 mae-means
    int l = (tid - 1) % 5;
    bool is_mae = (tid - 1) >= 5;
    float s = 0.0f;
#pragma unroll
    for (int t = 0; t < NT; ++t) s += is_mae ? sh_mae[t][l] : sh_ts[t][l];
    out[(is_mae ? 246 : 241) + l] = s * (1.0f / NT);
  }
}

extern "C" void kernel_launch(void* const* d_in, const int* in_sizes, int n_in,
                              void* d_out, int out_size, void* d_ws, size_t ws_size,
                              hipStream_t stream) {
  const float*         pred = (const float*)d_in[0];
  const float*         targ = (const float*)d_in[1];
  const unsigned char* mask = (const unsigned char*)d_in[2];
  float* part = (float*)d_ws;            // NBLK*NACC*4 = 320 KB of scratch
  float* out  = (float*)d_out;           // 251 floats

  met_partial<<<dim3(NBLK), dim3(TPB), 0, stream>>>(pred, targ, mask, part);
  met_final<<<dim3(1), dim3(NT * 32), 0, stream>>>(part, out);
}